// GaussianVectorQuantizer_12163347382689
// MI455X (gfx1250) — compile-verified
//
#include <hip/hip_runtime.h>
#include <math.h>

// GaussianVectorQuantizer for MI455X (gfx1250, wave32).
// Fused distance-GEMM (V_WMMA_F32_16X16X4_F32) + per-lane online softmax/argmax,
// single end-of-strip 16-lane merge (minimal ds_bpermute traffic in hot loop).
// HBM traffic ~70MB total; distance matrix never materialized.

typedef __attribute__((ext_vector_type(2))) float v2f;
typedef __attribute__((ext_vector_type(8))) float v8f;

#define BSZ   32
#define DIMZ  256
#define SP    1024            // W*H
#define NROW  (BSZ * SP)      // 32768
#define KCB   1024            // codebook entries
#define ROWS  32              // rows per block
#define NBLK  (NROW / ROWS)   // 1024 blocks
#define PITCH 258             // LDS pitch (floats): stride-2 banks -> conflict-free

// output layout (floats unless noted):
//   [0, 8388608)          z_to_decoder  [32,256,32,32]
//   [8388608]             loss
//   [8388609, 8421377)    indices (int32 bits)
//   [8421377]             perplexity
#define LOSS_OFF 8388608
#define IDX_OFF  8388609
#define PERP_OFF 8421377

// workspace layout (floats): [0,1024) cnorm | [1024,2048) pk | [2048,3072) pc
//                            [3072,4096) histogram (int32)   -- 16 KB total

__global__ __launch_bounds__(256) void vq_prep(const float* __restrict__ cb,
                                               float* __restrict__ ws) {
  const int t = blockIdx.x * 256 + threadIdx.x;  // 0..1023
  const float4* c4 = (const float4*)(cb + (size_t)t * DIMZ);
  float acc = 0.f;
  for (int i = 0; i < DIMZ / 4; ++i) {
    float4 v = c4[i];
    acc += v.x * v.x + v.y * v.y + v.z * v.z + v.w * v.w;
  }
  ws[t] = acc;                 // ||codebook_k||^2
  ws[1024 + t] = 0.f;          // kld_discrete partials
  ws[2048 + t] = 0.f;          // kld_continuous partials
  ((int*)ws)[3072 + t] = 0;    // argmax histogram
}

__global__ __launch_bounds__(256) void vq_main(const float* __restrict__ z,
                                               const float* __restrict__ varq,
                                               const float* __restrict__ cb,
                                               float* __restrict__ out,
                                               float* __restrict__ ws) {
  __shared__ float zs[ROWS][PITCH];
  __shared__ float wzn[ROWS];
  __shared__ float mS[ROWS][4], sS[ROWS][4], tS[ROWS][4];
  __shared__ int   aS[ROWS][4];
  __shared__ int   idxs[ROWS];
  __shared__ float wred[8];

  const int t    = threadIdx.x;
  const int lane = t & 31;
  const int w    = t >> 5;                 // wave id 0..7
  const int b    = blockIdx.x >> 5;        // batch index
  const int s0   = (blockIdx.x & 31) * ROWS;
  const float weight = 0.5f / fmaxf(varq[0], 1e-10f);

  const float* cnorm = ws;
  float* pk = ws + 1024;
  float* pc = ws + 2048;
  int*   hist = (int*)(ws + 3072);

  // ---- stage 32 z-rows into LDS (coalesced along s, conflict-free LDS writes)
  {
    const int r = t & 31;
    const int dbase = (t >> 5) * 32;
    const float* zp = z + (size_t)b * (DIMZ * SP) + s0 + r;
    for (int dd = 0; dd < 32; ++dd)
      zs[r][dbase + dd] = zp[(size_t)(dbase + dd) * SP];
  }
  __syncthreads();

  if (t < ROWS) {  // weight * ||z_row||^2
    float acc = 0.f;
    for (int d = 0; d < DIMZ; ++d) { float v = zs[t][d]; acc += v * v; }
    wzn[t] = weight * acc;
  }
  __syncthreads();

  // ---- fused GEMM + per-lane online softmax/argmax ----------------------
  const int g = w >> 2;            // row half (0/1)
  const int q = w & 3;             // code quarter
  const int rowBase = 16 * g;
  const int half = lane >> 4;      // ISA A/B fragment half
  const int ln16 = lane & 15;

  float wznl[8];
  for (int p = 0; p < 8; ++p) wznl[p] = wzn[rowBase + p + 8 * half];

  // lane-private stats over this lane's code subset (codes ≡ ln16 mod 16)
  float m[8], se[8], te[8]; int am[8];
  for (int p = 0; p < 8; ++p) { m[p] = -3.0e38f; se[p] = 0.f; te[p] = 0.f; am[p] = 0; }

  const float* zrow = &zs[rowBase + ln16][0];

  for (int tile = 0; tile < 16; ++tile) {
    const int code = q * 256 + tile * 16 + ln16;
    const float wcn = weight * cnorm[code];
    const float* crow = cb + (size_t)code * DIMZ;

    v8f acc = {0.f, 0.f, 0.f, 0.f, 0.f, 0.f, 0.f, 0.f};
    for (int k0 = 0; k0 < DIMZ; k0 += 4) {
      const int d = k0 + 2 * half;
      v2f a = *(const v2f*)(zrow + d);   // A 16x4 f32 fragment (ISA 7.12.2)
      v2f bf = *(const v2f*)(crow + d);  // B 4x16 f32 fragment
      acc = __builtin_amdgcn_wmma_f32_16x16x4_f32(
          false, a, false, bf, (short)0, acc, false, false);
    }
    // shuffle-free per-lane online update (one logit per row per lane)
    for (int p = 0; p < 8; ++p) {
      float l  = 2.f * weight * acc[p] - wznl[p] - wcn;  // logit = -w*dist
      float mn = fmaxf(m[p], l);
      float sc = __expf(m[p] - mn);
      float e  = __expf(l - mn);
      se[p] = se[p] * sc + e;
      te[p] = te[p] * sc + l * e;
      if (l > m[p]) am[p] = code;
      m[p] = mn;
    }
  }

  // single 16-lane merge of lane-private stats (value-then-index tiebreak)
  for (int p = 0; p < 8; ++p) {
    float bm = m[p]; int bi = am[p];
    float bs = se[p], bt = te[p];
    for (int off = 1; off <= 8; off <<= 1) {
      float om = __shfl_xor(bm, off, 32);
      int   oi = __shfl_xor(bi, off, 32);
      float os = __shfl_xor(bs, off, 32);
      float ot = __shfl_xor(bt, off, 32);
      float mn = fmaxf(bm, om);
      float s1 = __expf(bm - mn);
      float s2 = __expf(om - mn);
      bs = bs * s1 + os * s2;
      bt = bt * s1 + ot * s2;
      if (om > bm || (om == bm && oi < bi)) bi = oi;
      bm = mn;
    }
    m[p] = bm; se[p] = bs; te[p] = bt; am[p] = bi;
  }

  if (ln16 == 0) {  // stats replicated within 16-lane group
    for (int p = 0; p < 8; ++p) {
      const int row = rowBase + p + 8 * half;
      mS[row][q] = m[p]; sS[row][q] = se[p]; tS[row][q] = te[p]; aS[row][q] = am[p];
    }
  }
  __syncthreads();

  // ---- merge quarters per row: lse, sum p*logp, argmax, indices ---------
  if (t < ROWS) {
    float M = mS[t][0]; int idx = aS[t][0];
    for (int qq = 1; qq < 4; ++qq)
      if (mS[t][qq] > M) { M = mS[t][qq]; idx = aS[t][qq]; }
    float S = 0.f, T = 0.f;
    for (int qq = 0; qq < 4; ++qq) {
      float sc = __expf(mS[t][qq] - M);
      S += sS[t][qq] * sc;
      T += tS[t][qq] * sc;
    }
    float lse = M + __logf(S);
    float kld = T / S - lse;           // sum_k p*log p for this row
    idxs[t] = idx;
    ((int*)out)[IDX_OFF + (size_t)b * SP + s0 + t] = idx;
    atomicAdd(&hist[idx], 1);
    for (int off = 1; off <= 16; off <<= 1)   // wave0 owns all 32 rows
      kld += __shfl_xor(kld, off, 32);
    if (lane == 0) pk[blockIdx.x] = kld;
  }
  __syncthreads();

  // ---- gather + transpose write-back, kld_continuous partial ------------
  float csum = 0.f;
  {
    const int r = t & 31;
    const int dbase = (t >> 5) * 32;
    const float* crow = cb + (size_t)idxs[r] * DIMZ;
    float* op = out + (size_t)b * (DIMZ * SP) + s0 + r;
    for (int dd = 0; dd < 32; ++dd) {
      const int d = dbase + dd;
      float qv = crow[d];
      op[(size_t)d * SP] = qv;
      float df = zs[r][d] - qv;
      csum += df * df;
    }
  }
  for (int off = 1; off <= 16; off <<= 1) csum += __shfl_xor(csum, off, 32);
  if (lane == 0) wred[w] = csum;
  __syncthreads();
  if (t == 0) {
    float tot = 0.f;
    for (int i = 0; i < 8; ++i) tot += wred[i];
    pc[blockIdx.x] = weight * tot;
  }
}

__global__ __launch_bounds__(256) void vq_finalize(float* __restrict__ out,
                                                   const float* __restrict__ ws) {
  __shared__ float red[256];
  const int t = threadIdx.x;
  const float* pk = ws + 1024;
  const float* pc = ws + 2048;
  const int* hist = (const int*)(ws + 3072);

  float a = 0.f;
  for (int i = t; i < 1024; i += 256) a += pk[i] + pc[i];
  red[t] = a; __syncthreads();
  for (int s = 128; s > 0; s >>= 1) { if (t < s) red[t] += red[t + s]; __syncthreads(); }
  if (t == 0) out[LOSS_OFF] = red[0] / (float)BSZ;
  __syncthreads();

  float h = 0.f;
  for (int i = t; i < 1024; i += 256) {
    float avg = (float)hist[i] / (float)NROW;
    h += avg * __logf(avg + 1e-7f);
  }
  red[t] = h; __syncthreads();
  for (int s = 128; s > 0; s >>= 1) { if (t < s) red[t] += red[t + s]; __syncthreads(); }
  if (t == 0) out[PERP_OFF] = __expf(-red[0]);
}

extern "C" void kernel_launch(void* const* d_in, const int* in_sizes, int n_in,
                              void* d_out, int out_size, void* d_ws, size_t ws_size,
                              hipStream_t stream) {
  const float* z    = (const float*)d_in[0];  // [32,256,32,32]
  const float* varq = (const float*)d_in[1];  // [1]
  const float* cb   = (const float*)d_in[2];  // [1024,256]
  float* out = (float*)d_out;
  float* ws  = (float*)d_ws;                  // needs 16 KB

  vq_prep<<<4, 256, 0, stream>>>(cb, ws);
  vq_main<<<NBLK, 256, 0, stream>>>(z, varq, cb, out, ws);
  vq_finalize<<<1, 256, 0, stream>>>(out, ws);
}